// RIMLSAttNProcessor_81733227643073
// MI455X (gfx1250) — compile-verified
//
#include <hip/hip_runtime.h>
#include <hip/hip_bf16.h>

typedef __attribute__((ext_vector_type(2))) float v2f;
typedef __attribute__((ext_vector_type(8))) float v8f;

#define NQ    32768   // query points
#define NS    65536   // source points
#define KNN   30      // neighbors kept
#define QB    256     // queries per block (8 waves x 32 queries)
#define TS    32      // source tile width
#define TSP   36      // padded LDS row pitch (16B-aligned rows, float4 scans)
#define EPSF  1e-8f
#define SIGMA_N 0.8f
#define THRESH 1e-3f

// ---------------------------------------------------------------------------
// Kernel 1: normalize source normals (F.normalize(p=2, dim=1, eps=1e-8))
// ---------------------------------------------------------------------------
__global__ __launch_bounds__(256) void normals_kernel(const float* __restrict__ sn,
                                                      float* __restrict__ out) {
    int i = blockIdx.x * 256 + threadIdx.x;
    float x = sn[i * 3 + 0];
    float y = sn[i * 3 + 1];
    float z = sn[i * 3 + 2];
    float nrm = sqrtf(x * x + y * y + z * z);
    float inv = 1.0f / fmaxf(nrm, EPSF);
    out[i * 3 + 0] = x * inv;
    out[i * 3 + 1] = y * inv;
    out[i * 3 + 2] = z * inv;
}

// ---------------------------------------------------------------------------
// Kernel 2: exact KNN via V_WMMA_F32_16X16X4_F32 distance tiles + per-thread
// register top-30 selection (float4 LDS scans, quad quick-reject).
// ---------------------------------------------------------------------------
__global__ __launch_bounds__(256) void knn_kernel(const float* __restrict__ qp,
                                                  const float* __restrict__ sp,
                                                  float* __restrict__ out_d2,
                                                  int* __restrict__ out_idx) {
    __shared__ float4 q_lds[QB];        // x,y,z,|q|^2
    __shared__ float4 s_lds[TS];        // x,y,z,|s|^2
    __shared__ float  tile[QB * TSP];   // d^2 tile, thread-owns-row layout

    const int t     = threadIdx.x;
    const int lane  = t & 31;
    const int wave  = t >> 5;
    const int qbase = blockIdx.x * QB;
    const int wq    = wave * 32;        // this wave's first local query row
    const bool lo   = (lane < 16);

    // stage this block's 256 queries
    {
        float x = qp[(qbase + t) * 3 + 0];
        float y = qp[(qbase + t) * 3 + 1];
        float z = qp[(qbase + t) * 3 + 2];
        q_lds[t] = make_float4(x, y, z, x * x + y * y + z * z);
    }
    __syncthreads();

    // Hoisted A operands: 16x4 f32 A layout: VGPR0 = K0|K2, VGPR1 = K1|K3
    // (lanes 0-15 carry K0,K1 of row M=lane; lanes 16-31 carry K2,K3).
    // K layout: K0=x, K1=y, K2=z, K3=0 -> dot = q.s over 3 dims.
    v2f   a[2];
    float qq[2][8];
    const int half8 = lo ? 0 : 8;
#pragma unroll
    for (int mt = 0; mt < 2; ++mt) {
        float4 qv = q_lds[wq + mt * 16 + (lane & 15)];
        a[mt].x = lo ? qv.x : qv.z;
        a[mt].y = lo ? qv.y : 0.0f;
#pragma unroll
        for (int r = 0; r < 8; ++r)
            qq[mt][r] = q_lds[wq + mt * 16 + r + half8].w;
    }

    // per-thread running top-30 (unsorted, tracked max) — registers only
    float kd[KNN];
    int   ki[KNN];
#pragma unroll
    for (int j = 0; j < KNN; ++j) { kd[j] = 3.4e38f; ki[j] = 0; }
    float curmax  = 3.4e38f;
    int   maxslot = 0;

    const float4* rowp = reinterpret_cast<const float4*>(&tile[t * TSP]);

    for (int tile0 = 0; tile0 < NS; tile0 += TS) {
        __syncthreads();  // tile/s_lds reuse fence
        if (t < TS) {
            float x = sp[(tile0 + t) * 3 + 0];
            float y = sp[(tile0 + t) * 3 + 1];
            float z = sp[(tile0 + t) * 3 + 2];
            s_lds[t] = make_float4(x, y, z, x * x + y * y + z * z);
            // pull next tile toward the WGP while this tile is consumed
            if (tile0 + TS < NS)
                __builtin_prefetch(&sp[(size_t)(tile0 + TS + t) * 3], 0, 3);
        }
        __syncthreads();

        // ---- WMMA distance tile: each wave covers 32 queries x 32 sources ----
#pragma unroll
        for (int nt = 0; nt < TS / 16; ++nt) {
            float4 sv = s_lds[nt * 16 + (lane & 15)];   // one ds_load_b128
            v2f b;
            b.x = lo ? sv.x : sv.z;                     // branchless cndmask
            b.y = lo ? sv.y : 0.0f;
            float ssv = sv.w;                           // |s|^2 of column n
#pragma unroll
            for (int mt = 0; mt < 2; ++mt) {
                v8f c = {};
                c = __builtin_amdgcn_wmma_f32_16x16x4_f32(
                        /*neg_a=*/false, a[mt], /*neg_b=*/false, b,
                        /*c_mod=*/(short)0, c, /*reuse_a=*/false, /*reuse_b=*/false);
                // C layout: VGPR r holds M=r (lanes 0-15) / M=r+8 (lanes 16-31),
                // N = lane % 16.
                int rowbase = wq + mt * 16 + half8;
                int col     = nt * 16 + (lane & 15);
#pragma unroll
                for (int r = 0; r < 8; ++r) {
                    float d2 = fmaf(-2.0f, c[r], qq[mt][r] + ssv);
                    tile[(rowbase + r) * TSP + col] = d2;
                }
            }
        }
        __syncthreads();

        // ---- selection: thread t owns query (qbase + t), scans its row ----
        // NOTE: keep this loop rolled — the insert path is ~90 VALU ops and
        // must exist only once per quad element, not 32x (I-cache pressure).
#pragma unroll 1
        for (int c4 = 0; c4 < TS / 4; ++c4) {
            float4 d4 = rowp[c4];                        // ds_load_b128
            float m01 = fminf(d4.x, d4.y);
            float m23 = fminf(d4.z, d4.w);
            if (fminf(m01, m23) < curmax) {              // quad quick-reject
                float dv[4] = {d4.x, d4.y, d4.z, d4.w};
#pragma unroll
                for (int u = 0; u < 4; ++u) {
                    float d = dv[u];
                    if (d < curmax) {
                        int sid = tile0 + c4 * 4 + u;
#pragma unroll
                        for (int j = 0; j < KNN; ++j)
                            if (j == maxslot) { kd[j] = d; ki[j] = sid; }
                        curmax = kd[0]; maxslot = 0;
#pragma unroll
                        for (int j = 1; j < KNN; ++j)
                            if (kd[j] > curmax) { curmax = kd[j]; maxslot = j; }
                    }
                }
            }
        }
    }

#pragma unroll
    for (int j = 0; j < KNN; ++j) {
        out_d2[(qbase + t) * 32 + j]  = kd[j];
        out_idx[(qbase + t) * 32 + j] = ki[j];
    }
}

// ---------------------------------------------------------------------------
// Kernel 3: robust implicit MLS potential + gradient (3 masked iterations)
// ---------------------------------------------------------------------------
__global__ __launch_bounds__(256) void rimls_kernel(const float* __restrict__ qp,
                                                    const float* __restrict__ sp,
                                                    const float* __restrict__ nrm,
                                                    const float* __restrict__ kd,
                                                    const int* __restrict__ ki,
                                                    float* __restrict__ out) {
    const int q = blockIdx.x * 256 + threadIdx.x;
    const float qx = qp[q * 3 + 0];
    const float qy = qp[q * 3 + 1];
    const float qz = qp[q * 3 + 2];

    // bandwidth h = mean neighbor distance + eps
    float hs = 0.0f;
    for (int j = 0; j < KNN; ++j)
        hs += sqrtf(fmaxf(kd[q * 32 + j], 0.0f));
    const float h       = hs * (1.0f / (float)KNN) + EPSF;
    const float inv_h2  = 1.0f / (h * h);
    const float inv_sn2 = 1.0f / (SIGMA_N * SIGMA_N);

    float f = 0.0f, gx = 0.0f, gy = 0.0f, gz = 0.0f;
    bool  done = false;

    for (int it = 0; it < 3; ++it) {
        float sw = EPSF, swfx = 0.0f;
        float sgwfx_x = 0.0f, sgwfx_y = 0.0f, sgwfx_z = 0.0f;
        float swn_x = 0.0f, swn_y = 0.0f, swn_z = 0.0f;
        float sgw_x = 0.0f, sgw_y = 0.0f, sgw_z = 0.0f;

        for (int j = 0; j < KNN; ++j) {
            int   s  = ki[q * 32 + j];
            float px = qx - sp[s * 3 + 0];
            float py = qy - sp[s * 3 + 1];
            float pz = qz - sp[s * 3 + 2];
            float nx = nrm[s * 3 + 0];
            float ny = nrm[s * 3 + 1];
            float nz = nrm[s * 3 + 2];

            float fx = px * nx + py * ny + pz * nz;    // signed plane distance
            float r2 = px * px + py * py + pz * pz;
            float tt = fmaxf(1.0f - r2 * inv_h2, 0.0f);
            float t2 = tt * tt;
            float phi = t2 * t2;                       // t^4
            float gco = (-8.0f * inv_h2) * tt * t2;    // -8/h2 * t^3

            float alpha = 1.0f;
            if (it > 0) {
                float dx = nx - gx, dy = ny - gy, dz = nz - gz;
                alpha = expf(-(dx * dx + dy * dy + dz * dz) * inv_sn2);
            }
            float w  = alpha * phi;
            float gc = alpha * gco;     // gw = gc * px_vec

            sw      += w;
            swfx    += w * fx;
            sgwfx_x += gc * px * fx;  sgwfx_y += gc * py * fx;  sgwfx_z += gc * pz * fx;
            swn_x   += w * nx;        swn_y   += w * ny;        swn_z   += w * nz;
            sgw_x   += gc * px;       sgw_y   += gc * py;       sgw_z   += gc * pz;
        }

        float inv_sw = 1.0f / sw;
        float f_new  = swfx * inv_sw;
        float gnx = (sgwfx_x + swn_x - f_new * sgw_x) * inv_sw;
        float gny = (sgwfx_y + swn_y - f_new * sgw_y) * inv_sw;
        float gnz = (sgwfx_z + swn_z - f_new * sgw_z) * inv_sw;

        float delta = fabsf(f_new - f);
        if (!done) { f = f_new; gx = gnx; gy = gny; gz = gnz; }
        if (delta < THRESH) done = true;   // MIN_IT == 1 -> always armed
    }

    out[q]              = f;               // potential [N]
    out[NQ + q * 3 + 0] = gx;              // gradient [N,3]
    out[NQ + q * 3 + 1] = gy;
    out[NQ + q * 3 + 2] = gz;
}

// ---------------------------------------------------------------------------
extern "C" void kernel_launch(void* const* d_in, const int* in_sizes, int n_in,
                              void* d_out, int out_size, void* d_ws, size_t ws_size,
                              hipStream_t stream) {
    const float* qp = (const float*)d_in[0];   // query_points   [32768,3]
    const float* sp = (const float*)d_in[1];   // source_vertices[65536,3]
    const float* sn = (const float*)d_in[2];   // source_normals [65536,3]

    float* ws   = (float*)d_ws;
    float* nrm  = ws;                                             // 65536*3 floats
    float* kd   = ws + (size_t)NS * 3;                            // 32768*32 floats
    int*   kidx = (int*)(ws + (size_t)NS * 3 + (size_t)NQ * 32);  // 32768*32 ints
    float* out  = (float*)d_out;

    normals_kernel<<<NS / 256, 256, 0, stream>>>(sn, nrm);
    knn_kernel<<<NQ / QB, 256, 0, stream>>>(qp, sp, kd, kidx);
    rimls_kernel<<<NQ / 256, 256, 0, stream>>>(qp, sp, nrm, kd, kidx, out);
}